// FasterTensorProduct_7928509628863
// MI455X (gfx1250) — compile-verified
//
#include <hip/hip_runtime.h>

typedef float v2f __attribute__((ext_vector_type(2)));
typedef float v8f __attribute__((ext_vector_type(8)));

#define IN_DIM 176
#define OUT_DIM 176
#define WNUM 7936     // 80*64 + 96*16 + 48*16 + 32*16
#define SLICE 544     // 80 + 288 + 144 + 32
#define ZPAD  304     // shared zero arena (covers max masked-lane index 300)

__device__ __forceinline__ v8f wmma_f32(v2f a, v2f b, v8f c) {
  // D = A(16x4 f32) * B(4x16 f32) + C(16x16 f32)
  return __builtin_amdgcn_wmma_f32_16x16x4_f32(false, a, false, b, (short)0, c,
                                               false, false);
}

__device__ __forceinline__ float ldnt(const float* p) {
  return __builtin_nontemporal_load(p);   // single-use weight stream: TH=NT
}
__device__ __forceinline__ void stnt(float* p, float v) {
  __builtin_nontemporal_store(v, p);
}

__global__ __launch_bounds__(256) void ftp_kernel(const float* __restrict__ in_,
                                                  const float* __restrict__ sh,
                                                  const float* __restrict__ wgt,
                                                  float* __restrict__ out,
                                                  int nE) {
  __shared__ float smem[ZPAD + 8 * SLICE];
  const int tid  = threadIdx.x;
  const int lane = tid & 31;
  const int wv   = tid >> 5;
  const int e    = blockIdx.x * 8 + wv;

  // zero arena for masked A lanes (block-shared, written before the barrier)
  float* zar = smem;
  for (int i = tid; i < ZPAD; i += 256) zar[i] = 0.f;

  if (e < nE) {   // wave-uniform guard; EXEC full inside for WMMA
    float* sl = smem + ZPAD + wv * SLICE;
    // slice layout (floats): [0,80) out0e | [80,368) out1o (i-major,c-minor)
    //                        [368,512) out1e | [512,544) out0o
    float* aE  = sl;
    float* a1o = sl + 80;
    float* a1e = sl + 368;
    float* aO  = sl + 512;

    const float* pi = in_ + (size_t)e * IN_DIM;
    const float* ps = sh  + (size_t)e * 4;

    // prefetch this element's weight stream (256B cachelines)
    const float* pwp = wgt + (size_t)e * WNUM;
    #pragma unroll
    for (int j = 0; j < 4; ++j) {
      int off = lane * 64 + j * 2048;
      if (off < WNUM) __builtin_prefetch(pwp + off, 0, 0);
    }

    const float sh0 = ps[0], s1x = ps[1], s1y = ps[2], s1z = ps[3];
    const float inv_s2 = 0.70710678118654752440f;  // 1/sqrt(2)
    const float inv_s3 = 0.57735026918962576451f;  // 1/sqrt(3)

    // -------------- stage 1: build activation vectors in LDS --------------
    {
      int i = lane;                       // x0e, first half
      float v = pi[i];
      aE[i] = v * sh0;
      a1o[i*3+0] = v * s1x; a1o[i*3+1] = v * s1y; a1o[i*3+2] = v * s1z;
      i = lane + 32;                      // x0e, second half
      v = pi[i];
      aE[i] = v * sh0;
      a1o[i*3+0] = v * s1x; a1o[i*3+1] = v * s1y; a1o[i*3+2] = v * s1z;
    }
    if (lane < 16) {
      int v = lane;                       // x1o vector v
      float ax = pi[64 + 3*v + 0], ay = pi[64 + 3*v + 1], az = pi[64 + 3*v + 2];
      aE[64 + v] = (ax*s1x + ay*s1y + az*s1z) * inv_s3;
      a1o[(64+v)*3+0] = ax * sh0; a1o[(64+v)*3+1] = ay * sh0; a1o[(64+v)*3+2] = az * sh0;
      a1e[v*3+0] = (ay*s1z - az*s1y) * inv_s2;
      a1e[v*3+1] = (az*s1x - ax*s1z) * inv_s2;
      a1e[v*3+2] = (ax*s1y - ay*s1x) * inv_s2;
      float t = pi[160 + v];              // x0o
      aO[16 + v] = t * sh0;
      a1e[(32+v)*3+0] = t * s1x; a1e[(32+v)*3+1] = t * s1y; a1e[(32+v)*3+2] = t * s1z;
    } else {
      int v = lane - 16;                  // x1e vector v
      float bx = pi[112 + 3*v + 0], by = pi[112 + 3*v + 1], bz = pi[112 + 3*v + 2];
      aO[v] = (bx*s1x + by*s1y + bz*s1z) * inv_s3;
      a1e[(16+v)*3+0] = bx * sh0; a1e[(16+v)*3+1] = by * sh0; a1e[(16+v)*3+2] = bz * sh0;
      a1o[(80+v)*3+0] = (by*s1z - bz*s1y) * inv_s2;
      a1o[(80+v)*3+1] = (bz*s1x - bx*s1z) * inv_s2;
      a1o[(80+v)*3+2] = (bx*s1y - by*s1x) * inv_s2;
    }
  }
  __syncthreads();
  if (e >= nE) return;

  {
    float* sl = smem + ZPAD + wv * SLICE;
    const float* pw = wgt + (size_t)e * WNUM;
    float*       po = out + (size_t)e * OUT_DIM;

    // A layout for f32 16x16x4: lane m = lane&15 holds row M=m;
    // VGPR0 -> K = kb + 2*(lane>>4), VGPR1 -> K+1
    const int  m    = lane & 15;
    const int  kh2  = (lane >> 4) << 1;
    const bool row0 = (m == 0);
    const bool row3 = (m < 3);

    // per-lane A base pointers: one cndmask each, hoisted out of the loops;
    // inner loops use pure immediate offsets into LDS
    const float* baseE  = row0 ? (sl + kh2)             : (zar + kh2);
    const float* base1o = row3 ? (sl + 80  + kh2*3 + m) : (zar + kh2*3 + m);
    const float* base1e = row3 ? (sl + 368 + kh2*3 + m) : (zar + kh2*3 + m);
    const float* baseO  = row0 ? (sl + 512 + kh2)       : (zar + kh2);

    // -------------- GEMM '0e': (1x80) x (80x64), ldb=64 --------------
    {
      v8f c0 = {}, c1 = {}, c2 = {}, c3 = {};
      #pragma unroll
      for (int kb = 0; kb < 80; kb += 4) {
        v2f a = { baseE[kb], baseE[kb + 1] };        // ds_load_b64, imm offset
        const float* r = pw + (kb + kh2) * 64 + m;
        v2f b0 = { ldnt(r +  0), ldnt(r +  64) };
        v2f b1 = { ldnt(r + 16), ldnt(r +  80) };
        v2f b2 = { ldnt(r + 32), ldnt(r +  96) };
        v2f b3 = { ldnt(r + 48), ldnt(r + 112) };
        c0 = wmma_f32(a, b0, c0);
        c1 = wmma_f32(a, b1, c1);
        c2 = wmma_f32(a, b2, c2);
        c3 = wmma_f32(a, b3, c3);
      }
      const float rs = 0.11180339887498948f;  // 1/sqrt(80)
      if (lane < 16) {
        stnt(po + lane,      c0[0] * rs);
        stnt(po + 16 + lane, c1[0] * rs);
        stnt(po + 32 + lane, c2[0] * rs);
        stnt(po + 48 + lane, c3[0] * rs);
      }
    }

    // -------------- GEMM '1o': (3x96) x (96x16), ldb=16 --------------
    {
      const float* w = pw + 5120 + m;
      v8f c = {};
      #pragma unroll
      for (int kb = 0; kb < 96; kb += 4) {
        v2f a = { base1o[kb*3], base1o[kb*3 + 3] };
        int k0 = kb + kh2;
        v2f b = { ldnt(w + k0*16), ldnt(w + (k0+1)*16) };
        c = wmma_f32(a, b, c);
      }
      const float rs = 0.10206207261596575f;  // 1/sqrt(96)
      if (lane < 16) {
        stnt(po + 64 + lane*3 + 0, c[0] * rs);
        stnt(po + 64 + lane*3 + 1, c[1] * rs);
        stnt(po + 64 + lane*3 + 2, c[2] * rs);
      }
    }

    // -------------- GEMM '1e': (3x48) x (48x16), ldb=16 --------------
    {
      const float* w = pw + 6656 + m;
      v8f c = {};
      #pragma unroll
      for (int kb = 0; kb < 48; kb += 4) {
        v2f a = { base1e[kb*3], base1e[kb*3 + 3] };
        int k0 = kb + kh2;
        v2f b = { ldnt(w + k0*16), ldnt(w + (k0+1)*16) };
        c = wmma_f32(a, b, c);
      }
      const float rs = 0.14433756729740643f;  // 1/sqrt(48)
      if (lane < 16) {
        stnt(po + 112 + lane*3 + 0, c[0] * rs);
        stnt(po + 112 + lane*3 + 1, c[1] * rs);
        stnt(po + 112 + lane*3 + 2, c[2] * rs);
      }
    }

    // -------------- GEMM '0o': (1x32) x (32x16), ldb=16 --------------
    {
      const float* w = pw + 7424 + m;
      v8f c = {};
      #pragma unroll
      for (int kb = 0; kb < 32; kb += 4) {
        v2f a = { baseO[kb], baseO[kb + 1] };
        int k0 = kb + kh2;
        v2f b = { ldnt(w + k0*16), ldnt(w + (k0+1)*16) };
        c = wmma_f32(a, b, c);
      }
      const float rs = 0.17677669529663687f;  // 1/sqrt(32)
      if (lane < 16) stnt(po + 160 + lane, c[0] * rs);
    }
  }
}

extern "C" void kernel_launch(void* const* d_in, const int* in_sizes, int n_in,
                              void* d_out, int out_size, void* d_ws, size_t ws_size,
                              hipStream_t stream) {
  (void)n_in; (void)out_size; (void)d_ws; (void)ws_size;
  const float* in_ = (const float*)d_in[0];
  const float* sh  = (const float*)d_in[1];
  const float* wgt = (const float*)d_in[2];
  float* out = (float*)d_out;
  const int nE = in_sizes[0] / IN_DIM;       // E elements
  const int blocks = (nE + 7) / 8;           // 8 waves (elements) per block
  hipLaunchKernelGGL(ftp_kernel, dim3(blocks), dim3(256), 0, stream,
                     in_, sh, wgt, out, nE);
}